// PR2d_30004641530294
// MI455X (gfx1250) — compile-verified
//
#include <hip/hip_runtime.h>
#include <math.h>

// Pole-residue spectral operator for MI455X (gfx1250, wave32).
// All dense contractions (fwd DFT2, freq-mixing GEMM, inv DFT2) run on
// V_WMMA_F32_16X16X4_F32 (the CDNA5 fp32 matrix path). Hw is generated
// on the fly per frequency (never materialized: 537MB avoided).
// 64KB image tiles are staged into LDS by the Tensor Data Mover
// (tensor_load_to_lds + s_wait_tensorcnt), not by VGPR round trips.
// Workspace requirement: ~470 MB in d_ws.

#define B_  16
#define I_  64
#define O_  64
#define M1_ 8
#define M2_ 8
#define X_  128
#define Y_  128
#define NF  (X_*Y_)      // 16384 per-image elements / frequencies
#define IO_ (I_*O_)      // 4096

typedef float v2f __attribute__((ext_vector_type(2)));
typedef float v8f __attribute__((ext_vector_type(8)));
typedef unsigned int u32x4 __attribute__((ext_vector_type(4)));
typedef int i32x4 __attribute__((ext_vector_type(4)));
typedef int i32x8 __attribute__((ext_vector_type(8)));

#if defined(__gfx1250__) && __has_builtin(__builtin_amdgcn_tensor_load_to_lds) && \
    __has_builtin(__builtin_amdgcn_s_wait_tensorcnt)
#define USE_TDM 1
#else
#define USE_TDM 0
#endif

#if USE_TDM
#if __has_include(<hip/amd_detail/amd_gfx1250_TDM.h>)
// amdgpu-toolchain (therock headers): 6-arg builtin
#define TDM_CALL(g0, g1, z4, z8) __builtin_amdgcn_tensor_load_to_lds((g0), (g1), (z4), (z4), (z8), 0)
#else
// ROCm 7.2 lane: 5-arg builtin
#define TDM_CALL(g0, g1, z4, z8) __builtin_amdgcn_tensor_load_to_lds((g0), (g1), (z4), (z4), 0)
#endif

// ---- Tensor Data Mover: DMA one 128x128 f32 tile (global -> LDS) --------
// D# group0: [1:0]=count=1 | [63:32]=lds_addr | [120:64]=global_addr | [127:126]=type=2
// D# group1: data_size=2 (4B); tensor_dim0=tensor_dim1=128; tile_dim0=tile_dim1=128;
//            tensor_dim0_stride=128; tensor_dim1_stride=16384.
__device__ __forceinline__ void tdm_load_128x128_f32(const float* gptr, const float* lptr) {
  const unsigned long long ga = (unsigned long long)(uintptr_t)gptr;
  const unsigned lds = (unsigned)(uintptr_t)lptr;  // flat LDS addr: offset in [31:0]
  u32x4 g0;
  g0[0] = 1u;                                       // count=1, user descriptor
  g0[1] = lds;                                      // lds_addr (bytes)
  g0[2] = (unsigned)(ga & 0xFFFFFFFFull);           // global_addr[31:0]
  g0[3] = (unsigned)((ga >> 32) & 0x1FFFFFFull)     // global_addr[56:32]
          | 0x80000000u;                            // type=2 ("image")
  i32x8 g1;
  g1[0] = 0x20000;                                  // data_size=2 -> 4 bytes
  g1[1] = (int)(128u << 16);                        // tensor_dim0[15:0] @ bits63:48
  g1[2] = (int)(128u << 16);                        // tensor_dim1[15:0] @ bits95:80
  g1[3] = (int)(128u << 16);                        // tile_dim0 @ bits127:112
  g1[4] = 128;                                      // tile_dim1 (tile_dim2=0 -> 2D)
  g1[5] = 128;                                      // tensor_dim0_stride[31:0]
  g1[6] = (int)(16384u << 16);                      // tensor_dim1_stride[15:0] @ bits223:208
  g1[7] = 0;
  const i32x4 z4 = {0, 0, 0, 0};
  const i32x8 z8 = {0, 0, 0, 0, 0, 0, 0, 0};
  TDM_CALL(g0, g1, z4, z8);
  (void)z8;
}
#endif  // USE_TDM

__device__ __forceinline__ void stage_f4(float* dst, const float* src, int nFloats) {
  const float4* s = (const float4*)src;
  float4* d = (float4*)dst;
  for (int j = threadIdx.x; j < nFloats / 4; j += blockDim.x) d[j] = s[j];
}

// Stage three/four 64KB images into LDS. TDM path: wave 0 issues in-order
// DMA descriptors, waits TENSORcnt==0; __syncthreads() publishes to all waves.
__device__ __forceinline__ void stage_images3(float* d0, const float* s0,
                                              float* d1, const float* s1,
                                              float* d2, const float* s2) {
#if USE_TDM
  if (threadIdx.x < 32) {                 // wave-uniform branch: wave 0 only
    tdm_load_128x128_f32(s0, d0);
    tdm_load_128x128_f32(s1, d1);
    tdm_load_128x128_f32(s2, d2);
    __builtin_amdgcn_s_wait_tensorcnt(0);
  }
#else
  stage_f4(d0, s0, NF);
  stage_f4(d1, s1, NF);
  stage_f4(d2, s2, NF);
#endif
  __syncthreads();
}

__device__ __forceinline__ void stage_images4(float* d0, const float* s0,
                                              float* d1, const float* s1,
                                              float* d2, const float* s2,
                                              float* d3, const float* s3) {
#if USE_TDM
  if (threadIdx.x < 32) {
    tdm_load_128x128_f32(s0, d0);
    tdm_load_128x128_f32(s1, d1);
    tdm_load_128x128_f32(s2, d2);
    tdm_load_128x128_f32(s3, d3);
    __builtin_amdgcn_s_wait_tensorcnt(0);
  }
#else
  stage_f4(d0, s0, NF);
  stage_f4(d1, s1, NF);
  stage_f4(d2, s2, NF);
  stage_f4(d3, s3, NF);
#endif
  __syncthreads();
}

// ---------------- WMMA helpers (wave32, 16x16 tiles, K steps of 4) -------
// A: MxK row-major (lda), B: KxN row-major (ldb), both expected in LDS.
// A frag layout: lanes0-15 row M=l16 hold K={k0,k0+1}; lanes16-31 K={k0+2,k0+3}.
template<int K>
__device__ __forceinline__ v8f wmma_tile_acc(const float* A, int lda,
                                             const float* Bm, int ldb,
                                             int mBase, int nBase, v8f acc) {
  const int lane = threadIdx.x & 31;
  const int l16  = lane & 15;
  const int h    = lane >> 4;
  const float* arow = A + (mBase + l16) * lda;
  #pragma unroll
  for (int k0 = 0; k0 < K; k0 += 4) {
    const int ka = k0 + 2 * h;
    v2f a, b;
    a.x = arow[ka];
    a.y = arow[ka + 1];
    b.x = Bm[ka * ldb + nBase + l16];
    b.y = Bm[(ka + 1) * ldb + nBase + l16];
    acc = __builtin_amdgcn_wmma_f32_16x16x4_f32(
        /*neg_a=*/false, a, /*neg_b=*/false, b,
        /*c_mod=*/(short)0, acc, /*reuse_a=*/false, /*reuse_b=*/false);
  }
  return acc;
}

// C/D layout: VGPR r -> row M = mBase + r + 8*(lane>=16), col N = nBase + lane%16
__device__ __forceinline__ void store_tile(float* C, int ldc, int mBase, int nBase, v8f acc) {
  const int lane = threadIdx.x & 31;
  const int l16  = lane & 15;
  const int h    = lane >> 4;
  #pragma unroll
  for (int r = 0; r < 8; ++r)
    C[(size_t)(mBase + r + 8 * h) * ldc + nBase + l16] = acc[r];
}

// ---------------- K0: DFT matrices (symmetric: W[y][p]==W[p][y]) ---------
// Fwd: exp(-2pi*i*yp/128). Inv: exp(+2pi*i*yp/128)/128. Exact mod-128 arg
// reduction keeps sinf/cosf accurate.
__global__ void k_build_dft(float* Fre, float* Fim, float* Vre, float* Vim) {
  const int tid = blockIdx.x * 256 + threadIdx.x;   // 16384
  const int y = tid >> 7, p = tid & 127;
  const int m = (y * p) & 127;
  const float ang = 6.283185307179586f * (float)m / 128.0f;
  const float c = cosf(ang), s = sinf(ang);
  Fre[tid] = c;
  Fim[tid] = -s;
  Vre[tid] = c * (1.0f / 128.0f);
  Vim[tid] = s * (1.0f / 128.0f);
}

// ---------------- K1: forward DFT along Y (real input) -------------------
// A1[x][p] = sum_y x[x,y] * F[y][p].  One image per workgroup; LDS = 192KB.
__global__ void __launch_bounds__(256) k_fwd_row(const float* __restrict__ x,
    const float* __restrict__ Fre, const float* __restrict__ Fim,
    float* __restrict__ A1re, float* __restrict__ A1im) {
  extern __shared__ float sm[];
  float* sX = sm; float* sBre = sm + NF; float* sBim = sm + 2 * NF;
  const int img = blockIdx.x;                       // b*I + i
  stage_images3(sX, x + (size_t)img * NF, sBre, Fre, sBim, Fim);
  const int wave = threadIdx.x >> 5;
  const v8f z = {};
  for (int j = 0; j < 8; ++j) {
    const int tile = wave * 8 + j;
    const int mB = (tile >> 3) * 16, nB = (tile & 7) * 16;
    v8f cr = wmma_tile_acc<128>(sX, 128, sBre, 128, mB, nB, z);
    v8f ci = wmma_tile_acc<128>(sX, 128, sBim, 128, mB, nB, z);
    store_tile(A1re + (size_t)img * NF, 128, mB, nB, cr);
    store_tile(A1im + (size_t)img * NF, 128, mB, nB, ci);
  }
}

// ---------------- K2: forward DFT along X, write alpha freq-major --------
// alpha[p][q][b][i] so K4's per-frequency A-tile (16x64) is contiguous.
__global__ void __launch_bounds__(256) k_fwd_col(const float* __restrict__ A1re,
    const float* __restrict__ A1im,
    const float* __restrict__ Fre, const float* __restrict__ Fim,
    float* __restrict__ aRe, float* __restrict__ aIm) {
  extern __shared__ float sm[];
  float* sWre = sm; float* sWim = sm + NF;
  float* sBre = sm + 2 * NF; float* sBim = sm + 3 * NF;
  const int img = blockIdx.x;
  const int b = img >> 6, i = img & 63;
  stage_images4(sWre, Fre, sWim, Fim,
                sBre, A1re + (size_t)img * NF, sBim, A1im + (size_t)img * NF);
  const int wave = threadIdx.x >> 5;
  const int lane = threadIdx.x & 31, l16 = lane & 15, h = lane >> 4;
  const v8f z = {};
  for (int j = 0; j < 8; ++j) {
    const int tile = wave * 8 + j;
    const int mB = (tile >> 3) * 16, nB = (tile & 7) * 16;
    v8f rr = wmma_tile_acc<128>(sWre, 128, sBre, 128, mB, nB, z);
    v8f ii = wmma_tile_acc<128>(sWim, 128, sBim, 128, mB, nB, z);
    v8f ri = wmma_tile_acc<128>(sWre, 128, sBim, 128, mB, nB, z);
    v8f ir = wmma_tile_acc<128>(sWim, 128, sBre, 128, mB, nB, z);
    #pragma unroll
    for (int r = 0; r < 8; ++r) {
      const int p = mB + r + 8 * h;
      const int q = nB + l16;
      const size_t idx = (size_t)(p * Y_ + q) * (B_ * I_) + b * I_ + i;
      aRe[idx] = rr[r] - ii[r];   // complex combine (f32 WMMA has no A/B neg)
      aIm[idx] = ri[r] + ir[r];
    }
  }
}

// ---------------- K3: term1 and G = residue contracted with term2 --------
// lam = i*w  =>  1/(lam - pole) = (a - i*bb)/(a^2+bb^2), a=-pole.re, bb=w-pole.im
// Layouts: t1[p][io][n], G[q][io][n]  (contiguous slices for K4).
__global__ void k_terms(const float2* __restrict__ wp1, const float2* __restrict__ wp2,
                        const float2* __restrict__ wres,
                        float* __restrict__ t1re, float* __restrict__ t1im,
                        float* __restrict__ Gre, float* __restrict__ Gim) {
  const int tid = blockIdx.x * 256 + threadIdx.x;   // 4,194,304 = 128*4096*8
  const int n  = tid & 7;
  const int io = (tid >> 3) & (IO_ - 1);
  const int s  = tid >> 15;                          // freq index (p or q)
  const float w = 6.283185307179586f * (float)((s < 64) ? s : s - 128) / 128.0f;
  // term1(i,o,n,p)
  {
    const float2 p1 = wp1[io * M1_ + n];
    const float a = -p1.x, bb = w - p1.y;
    const float d = a * a + bb * bb;
    t1re[tid] = a / d;
    t1im[tid] = -bb / d;
  }
  // G(i,o,n,q) = sum_m res(i,o,n,m) * term2(i,o,m,q)
  {
    float gr = 0.f, gi = 0.f;
    #pragma unroll
    for (int m = 0; m < M2_; ++m) {
      const float2 p2 = wp2[io * M2_ + m];
      const float a = -p2.x, bb = w - p2.y;
      const float d = a * a + bb * bb;
      const float tr = a / d, ti = -bb / d;
      const float2 rs = wres[io * (M1_ * M2_) + n * M2_ + m];
      gr += rs.x * tr - rs.y * ti;
      gi += rs.x * ti + rs.y * tr;
    }
    Gre[tid] = gr;
    Gim[tid] = gi;
  }
}

// ---------------- K4: per-frequency mixing GEMM --------------------------
// Hw[i][o] = sum_n G[q][io][n]*t1[p][io][n]  (VALU, into LDS), then
// out_fft[16x64] = alpha[16x64] * Hw[64x64] complex, on WMMA (4 real chains).
__global__ void __launch_bounds__(256) k_mix(const float* __restrict__ aRe,
    const float* __restrict__ aIm,
    const float* __restrict__ t1re, const float* __restrict__ t1im,
    const float* __restrict__ Gre, const float* __restrict__ Gim,
    float* __restrict__ oFre, float* __restrict__ oFim) {
  __shared__ float sHr[IO_], sHi[IO_], sAr[B_ * I_], sAi[B_ * I_];
  const int fq = blockIdx.x;
  const int p = fq >> 7, q = fq & 127;
  const int tid = threadIdx.x;
  // stage A (16x64 complex, contiguous)
  ((float4*)sAr)[tid] = ((const float4*)(aRe + (size_t)fq * (B_ * I_)))[tid];
  ((float4*)sAi)[tid] = ((const float4*)(aIm + (size_t)fq * (B_ * I_)))[tid];
  // generate Hw: 16 consecutive io per thread (512B-contiguous slice reads)
  {
    const float* t1r = t1re + (size_t)p * (IO_ * M1_);
    const float* t1i = t1im + (size_t)p * (IO_ * M1_);
    const float* gr_ = Gre + (size_t)q * (IO_ * M1_);
    const float* gi_ = Gim + (size_t)q * (IO_ * M1_);
    for (int j = 0; j < 16; ++j) {
      const int io = tid * 16 + j;
      const int base = io * M1_;
      float hr = 0.f, hi = 0.f;
      #pragma unroll
      for (int nn = 0; nn < M1_; ++nn) {
        const float ar = t1r[base + nn], ai = t1i[base + nn];
        const float gr = gr_[base + nn], gi = gi_[base + nn];
        hr += gr * ar - gi * ai;
        hi += gr * ai + gi * ar;
      }
      sHr[io] = hr;
      sHi[io] = hi;
    }
  }
  __syncthreads();
  const int wave = tid >> 5;
  const int nB = (wave & 3) * 16;
  const int part = wave >> 2;                        // 0: real out, 1: imag out
  const int lane = tid & 31, l16 = lane & 15, h = lane >> 4;
  const v8f z = {};
  if (part == 0) {
    v8f a1 = wmma_tile_acc<64>(sAr, 64, sHr, 64, 0, nB, z);
    v8f a2 = wmma_tile_acc<64>(sAi, 64, sHi, 64, 0, nB, z);
    #pragma unroll
    for (int r = 0; r < 8; ++r) {
      const int b = r + 8 * h, o = nB + l16;
      oFre[(size_t)(b * O_ + o) * NF + fq] = a1[r] - a2[r];
    }
  } else {
    v8f a1 = wmma_tile_acc<64>(sAr, 64, sHi, 64, 0, nB, z);
    v8f a2 = wmma_tile_acc<64>(sAi, 64, sHr, 64, 0, nB, z);
    #pragma unroll
    for (int r = 0; r < 8; ++r) {
      const int b = r + 8 * h, o = nB + l16;
      oFim[(size_t)(b * O_ + o) * NF + fq] = a1[r] + a2[r];
    }
  }
}

// ---------------- K5: inverse DFT along X --------------------------------
// T[k][q] = sum_p V[k][p]*oF[p][q], per image (b,o). LDS = 256KB.
__global__ void __launch_bounds__(256) k_inv_col(const float* __restrict__ oFre,
    const float* __restrict__ oFim,
    const float* __restrict__ Vre, const float* __restrict__ Vim,
    float* __restrict__ Tre, float* __restrict__ Tim) {
  extern __shared__ float sm[];
  float* sWre = sm; float* sWim = sm + NF;
  float* sBre = sm + 2 * NF; float* sBim = sm + 3 * NF;
  const int img = blockIdx.x;                        // b*O + o
  stage_images4(sWre, Vre, sWim, Vim,
                sBre, oFre + (size_t)img * NF, sBim, oFim + (size_t)img * NF);
  const int wave = threadIdx.x >> 5;
  const int lane = threadIdx.x & 31, l16 = lane & 15, h = lane >> 4;
  const v8f z = {};
  for (int j = 0; j < 8; ++j) {
    const int tile = wave * 8 + j;
    const int mB = (tile >> 3) * 16, nB = (tile & 7) * 16;
    v8f rr = wmma_tile_acc<128>(sWre, 128, sBre, 128, mB, nB, z);
    v8f ii = wmma_tile_acc<128>(sWim, 128, sBim, 128, mB, nB, z);
    v8f ri = wmma_tile_acc<128>(sWre, 128, sBim, 128, mB, nB, z);
    v8f ir = wmma_tile_acc<128>(sWim, 128, sBre, 128, mB, nB, z);
    #pragma unroll
    for (int r = 0; r < 8; ++r) {
      const size_t idx = (size_t)img * NF + (size_t)(mB + r + 8 * h) * 128 + nB + l16;
      Tre[idx] = rr[r] - ii[r];
      Tim[idx] = ri[r] + ir[r];
    }
  }
}

// ---------------- K6: inverse DFT along Y, real part only ----------------
// out[x][y] = Re( sum_p T[x][p] * V[p][y] ) = Tre*Vre - Tim*Vim
__global__ void __launch_bounds__(256) k_inv_row(const float* __restrict__ Tre,
    const float* __restrict__ Tim,
    const float* __restrict__ Vre, const float* __restrict__ Vim,
    float* __restrict__ out) {
  extern __shared__ float sm[];
  float* sAre = sm; float* sAim = sm + NF;
  float* sWre = sm + 2 * NF; float* sWim = sm + 3 * NF;
  const int img = blockIdx.x;                        // b*O + o
  stage_images4(sAre, Tre + (size_t)img * NF, sAim, Tim + (size_t)img * NF,
                sWre, Vre, sWim, Vim);
  const int wave = threadIdx.x >> 5;
  const int lane = threadIdx.x & 31, l16 = lane & 15, h = lane >> 4;
  const v8f z = {};
  for (int j = 0; j < 8; ++j) {
    const int tile = wave * 8 + j;
    const int mB = (tile >> 3) * 16, nB = (tile & 7) * 16;
    v8f rr = wmma_tile_acc<128>(sAre, 128, sWre, 128, mB, nB, z);
    v8f ii = wmma_tile_acc<128>(sAim, 128, sWim, 128, mB, nB, z);
    #pragma unroll
    for (int r = 0; r < 8; ++r) {
      out[(size_t)img * NF + (size_t)(mB + r + 8 * h) * 128 + nB + l16] = rr[r] - ii[r];
    }
  }
}

// ---------------- host launcher ------------------------------------------
extern "C" void kernel_launch(void* const* d_in, const int* in_sizes, int n_in,
                              void* d_out, int out_size, void* d_ws, size_t ws_size,
                              hipStream_t stream) {
  const float*  x    = (const float*)d_in[0];
  const float2* wp1  = (const float2*)d_in[1];   // complex64 -> float2
  const float2* wp2  = (const float2*)d_in[2];
  const float2* wres = (const float2*)d_in[3];
  float* out = (float*)d_out;
  float* w = (float*)d_ws;

  // workspace layout (floats)
  float* Fre  = w;                     // 16384
  float* Fim  = Fre  + 16384;
  float* Vre  = Fim  + 16384;
  float* Vim  = Vre  + 16384;
  float* t1re = Vim  + 16384;          // 4,194,304 each
  float* t1im = t1re + 4194304;
  float* Gre  = t1im + 4194304;
  float* Gim  = Gre  + 4194304;
  float* A1re = Gim  + 4194304;        // 16,777,216 each
  float* A1im = A1re + 16777216;
  float* aRe  = A1im + 16777216;
  float* aIm  = aRe  + 16777216;
  float* oFre = aIm  + 16777216;
  float* oFim = oFre + 16777216;
  float* Tre  = A1re;                  // reuse fwd-row temp for inv-col temp
  float* Tim  = A1im;

  // allow >64KB dynamic LDS (WGP has 320KB)
  (void)hipFuncSetAttribute((const void*)k_fwd_row, hipFuncAttributeMaxDynamicSharedMemorySize, 196608);
  (void)hipFuncSetAttribute((const void*)k_fwd_col, hipFuncAttributeMaxDynamicSharedMemorySize, 262144);
  (void)hipFuncSetAttribute((const void*)k_inv_col, hipFuncAttributeMaxDynamicSharedMemorySize, 262144);
  (void)hipFuncSetAttribute((const void*)k_inv_row, hipFuncAttributeMaxDynamicSharedMemorySize, 262144);

  k_build_dft<<<64, 256, 0, stream>>>(Fre, Fim, Vre, Vim);
  k_fwd_row<<<B_ * I_, 256, 196608, stream>>>(x, Fre, Fim, A1re, A1im);
  k_fwd_col<<<B_ * I_, 256, 262144, stream>>>(A1re, A1im, Fre, Fim, aRe, aIm);
  k_terms<<<16384, 256, 0, stream>>>(wp1, wp2, wres, t1re, t1im, Gre, Gim);
  k_mix<<<NF, 256, 0, stream>>>(aRe, aIm, t1re, t1im, Gre, Gim, oFre, oFim);
  k_inv_col<<<B_ * O_, 256, 262144, stream>>>(oFre, oFim, Vre, Vim, Tre, Tim);
  k_inv_row<<<B_ * O_, 256, 262144, stream>>>(Tre, Tim, Vre, Vim, out);
}